// EfficientDetWithFixedOutputs_43336220016742
// MI455X (gfx1250) — compile-verified
//
#include <hip/hip_runtime.h>
#include <cmath>

#define TOPK   4096
#define MAXDET 100
#define THR    0.5f
#define NMSTHR 0.5f
#define CAP    8192
#define NBINS  65536

typedef float v2f __attribute__((ext_vector_type(2)));
typedef float v8f __attribute__((ext_vector_type(8)));

// ---------------- utility ----------------
__global__ void zero_u32(uint32_t* p, int n) {
  int i = blockIdx.x * blockDim.x + threadIdx.x;
  if (i < n) p[i] = 0u;
}

// ---------------- K1: per-anchor class max/argmax + box decode (async LDS staging) ----
__global__ __launch_bounds__(128)
void k1_score_decode(const float* __restrict__ cls,
                     const float* __restrict__ reg,
                     const float* __restrict__ anc,
                     float* __restrict__ oBoxes,
                     float* __restrict__ oCand,
                     int* __restrict__ oCls,
                     int N, int C, float W, float H)
{
  __shared__ __align__(16) float sh[128 * 124];
  const int lt = threadIdx.x;
  const int t  = blockIdx.x * 128 + lt;
  const int stride = ((C + 3) >> 2) << 2;   // dwords, keeps rows 16B aligned

  if (t < N) {
    uint64_t ga = (uint64_t)(uintptr_t)(cls + (size_t)t * C);
    uint32_t la = (uint32_t)(uintptr_t)(&sh[(size_t)lt * stride]);
    int bytes = C * 4, k = 0;
    for (; k + 16 <= bytes; k += 16)
      asm volatile("global_load_async_to_lds_b128 %0, %1, off"
                   :: "v"(la + (uint32_t)k), "v"(ga + (uint64_t)k) : "memory");
    for (; k + 8 <= bytes; k += 8)
      asm volatile("global_load_async_to_lds_b64 %0, %1, off"
                   :: "v"(la + (uint32_t)k), "v"(ga + (uint64_t)k) : "memory");
    for (; k + 4 <= bytes; k += 4)
      asm volatile("global_load_async_to_lds_b32 %0, %1, off"
                   :: "v"(la + (uint32_t)k), "v"(ga + (uint64_t)k) : "memory");
  }
  asm volatile("s_wait_asynccnt 0" ::: "memory");
  __syncthreads();
  if (t >= N) return;

  const float* row = &sh[(size_t)lt * stride];
  float best = row[0]; int bi = 0;
  for (int c = 1; c < C; ++c) { float v = row[c]; if (v > best) { best = v; bi = c; } }

  const float4 A4 = ((const float4*)anc)[t];  // y1,x1,y2,x2
  const float4 R4 = ((const float4*)reg)[t];  // dy,dx,dh,dw
  float ya = (A4.x + A4.z) * 0.5f, xa = (A4.y + A4.w) * 0.5f;
  float ha = A4.z - A4.x,          wa = A4.w - A4.y;
  float w  = expf(R4.w) * wa;
  float h  = expf(R4.z) * ha;
  float yc = R4.x * ha + ya, xc = R4.y * wa + xa;
  float x1 = fmaxf(xc - w * 0.5f, 0.f);
  float y1 = fmaxf(yc - h * 0.5f, 0.f);
  float x2 = fminf(xc + w * 0.5f, W - 1.f);
  float y2 = fminf(yc + h * 0.5f, H - 1.f);
  ((float4*)oBoxes)[t] = make_float4(x1, y1, x2, y2);
  oCand[t] = best;
  oCls[t]  = bi;
}

// ---------------- K2a: 65536-bin histogram of sortable keys ----------------
__global__ void k2a_hist(const float* __restrict__ cand, uint32_t* __restrict__ hist, int N) {
  int i = blockIdx.x * blockDim.x + threadIdx.x;
  if (i >= N) return;
  float s = cand[i];
  uint32_t key = (s > THR) ? __float_as_uint(s) : 0u;   // positive floats: bits monotonic
  atomicAdd(&hist[key >> 16], 1u);
}

// ---------------- K2b: find threshold bucket (suffix count >= TOPK) ----------------
__global__ __launch_bounds__(256)
void k2b_thresh(const uint32_t* __restrict__ hist, uint32_t* __restrict__ scal) {
  __shared__ uint32_t csum[256];
  int ci = threadIdx.x;
  uint32_t s = 0;
  for (int b = 0; b < 256; ++b) s += hist[ci * 256 + b];
  csum[ci] = s;
  __syncthreads();
  if (ci == 0) {
    uint32_t acc = 0, bstar = 1;
    for (int c = 255; c >= 0; --c) {
      if (acc + csum[c] >= (uint32_t)TOPK) {
        int lo = (c == 0) ? 1 : c * 256;           // bucket 0 = invalid keys, excluded
        for (int b = c * 256 + 255; b >= lo; --b) {
          acc += hist[b];
          if (acc >= (uint32_t)TOPK) { bstar = (uint32_t)b; goto done; }
        }
      } else {
        acc += csum[c];
      }
    }
done:
    scal[0] = bstar;
  }
}

// ---------------- K2c: collect candidates with bucket >= bstar ----------------
__global__ void k2c_collect(const float* __restrict__ cand, const uint32_t* __restrict__ scal,
                            uint32_t* __restrict__ counter,
                            uint32_t* __restrict__ selKey, int* __restrict__ selIdx, int N) {
  int i = blockIdx.x * blockDim.x + threadIdx.x;
  if (i >= N) return;
  float s = cand[i];
  uint32_t key = (s > THR) ? __float_as_uint(s) : 0u;
  if ((key >> 16) >= scal[0]) {
    uint32_t pos = atomicAdd(counter, 1u);
    if (pos < (uint32_t)CAP) { selKey[pos] = key; selIdx[pos] = i; }
  }
}

// ---------------- K2d-init: sentinel-fill sorted arrays ----------------
__global__ void k2d_init(float* sScore, float* sBox, int* sCls) {
  int i = blockIdx.x * blockDim.x + threadIdx.x;
  if (i < TOPK) {
    sScore[i] = -1.0f;
    sCls[i] = 0;
    ((float4*)sBox)[i] = make_float4(0.f, 0.f, 0.f, 0.f);
  }
}

// ---------------- K2d: O(M^2) stable rank sort (key desc, idx asc) ----------------
__global__ __launch_bounds__(256)
void k2d_rank(const uint32_t* __restrict__ selKey, const int* __restrict__ selIdx,
              const uint32_t* __restrict__ scal,
              const float* __restrict__ boxes, const int* __restrict__ cls,
              float* __restrict__ sScore, float* __restrict__ sBox, int* __restrict__ sCls) {
  __shared__ uint32_t kBuf[2048];
  __shared__ int      iBuf[2048];
  const int M = (int)(scal[1] < (uint32_t)CAP ? scal[1] : (uint32_t)CAP);
  const int g = blockIdx.x * 256 + threadIdx.x;
  const bool act = g < M;
  uint32_t myKey = 0; int myIdx = 0;
  if (act) { myKey = selKey[g]; myIdx = selIdx[g]; }
  int rank = 0;
  for (int base = 0; base < M; base += 2048) {
    int cnt = (M - base < 2048) ? (M - base) : 2048;
    __syncthreads();
    for (int j = threadIdx.x; j < cnt; j += 256) { kBuf[j] = selKey[base + j]; iBuf[j] = selIdx[base + j]; }
    __syncthreads();
    if (act) {
      for (int j = 0; j < cnt; ++j) {
        uint32_t kj = kBuf[j];
        rank += (int)((kj > myKey) || (kj == myKey && iBuf[j] < myIdx));
      }
    }
  }
  if (act && rank < TOPK) {
    sScore[rank] = __uint_as_float(myKey);
    ((float4*)sBox)[rank] = ((const float4*)boxes)[myIdx];
    sCls[rank] = cls[myIdx];
  }
}

// ---------------- K3a: suppression bit-matrix; WMMA computes area[i]+area[j] tile ----
// One wave per 16-row i-tile sweeps all 256 j-tiles. No LDS, no barriers:
// i-tile boxes live in registers; ballot results are wave-uniform and sliced
// per-lane with a cndmask chain.
__global__ __launch_bounds__(32)
void k3a_supmat(const float* __restrict__ sBox, uint32_t* __restrict__ sup) {
  const int l     = threadIdx.x;
  const int it    = blockIdx.x;          // i-tile, 16 rows
  const int n     = l & 15;
  const int mbase = (l >> 4) << 3;
  const bool lo16 = (l < 16);

  // register-cache the 8 i-rows this lane evaluates (loop-invariant)
  float4 bi[8];
#pragma unroll
  for (int v = 0; v < 8; ++v)
    bi[v] = ((const float4*)sBox)[it * 16 + mbase + v];

  // A operand (loop-invariant): lanes 0-15 hold K=0 -> area_i[row l], K=1 -> 1.
  float4 rowl = ((const float4*)sBox)[it * 16 + n];
  float areai = fmaxf(rowl.z - rowl.x, 0.f) * fmaxf(rowl.w - rowl.y, 0.f);
  v2f A;
  A.x = lo16 ? areai : 0.f;
  A.y = lo16 ? 1.f   : 0.f;

  uint32_t acc = 0;
  for (int jt = 0; jt < TOPK / 16; ++jt) {
    float4 bj = ((const float4*)sBox)[jt * 16 + n];
    float areaj = fmaxf(bj.z - bj.x, 0.f) * fmaxf(bj.w - bj.y, 0.f);
    // B operand: lanes 0-15 hold K=0 row -> 1, K=1 row -> area_j[col l].
    v2f B;
    B.x = lo16 ? 1.f   : 0.f;
    B.y = lo16 ? areaj : 0.f;
    v8f Cz = {};
    // D[m][n] = area_i[m] + area_j[n]
    v8f D = __builtin_amdgcn_wmma_f32_16x16x4_f32(false, A, false, B, (short)0, Cz, false, false);

    uint32_t bits16 = 0;
#pragma unroll
    for (int v = 0; v < 8; ++v) {
      float ix1 = fmaxf(bi[v].x, bj.x);
      float iy1 = fmaxf(bi[v].y, bj.y);
      float ix2 = fminf(bi[v].z, bj.z);
      float iy2 = fminf(bi[v].w, bj.w);
      float inter = fmaxf(ix2 - ix1, 0.f) * fmaxf(iy2 - iy1, 0.f);
      float uni = fmaxf(D[v] - inter, 1e-8f);
      bool hit = (inter > NMSTHR * uni) && (jt * 16 + n > it * 16 + mbase + v);
      uint32_t bal = (uint32_t)__ballot((int)hit);  // uniform: [15:0]=row v, [31:16]=row v+8
      bits16 = (l == v)     ? (bal & 0xFFFFu) : bits16;
      bits16 = (l == v + 8) ? (bal >> 16)     : bits16;
    }
    acc |= bits16 << ((jt & 1) * 16);
    if (jt & 1) {
      if (l < 16)
        sup[(size_t)(it * 16 + l) * (TOPK / 32) + (jt >> 1)] = acc;
      acc = 0;
    }
  }
}

// ---------------- K3b: sequential greedy NMS scan + output scatter ----------------
__global__ __launch_bounds__(128)
void k3b_nms(const uint32_t* __restrict__ sup,
             const float* __restrict__ sScore, const float* __restrict__ sBox,
             const int* __restrict__ sCls, float* __restrict__ out) {
  __shared__ uint32_t keep[128];
  __shared__ uint32_t baseCnt[129];
  const int t = threadIdx.x;
  uint32_t w = 0;
  for (int b = 0; b < 32; ++b)
    w |= (sScore[t * 32 + b] > THR) ? (1u << b) : 0u;
  keep[t] = w;
  __syncthreads();
  for (int i = 0; i < TOPK; ++i) {
    uint32_t ki = (keep[i >> 5] >> (i & 31)) & 1u;
    __syncthreads();
    if (ki) keep[t] &= ~sup[(size_t)i * (TOPK / 32) + t];
    __syncthreads();
  }
  uint32_t cw = keep[t];
  baseCnt[t] = (uint32_t)__popc(cw);
  __syncthreads();
  if (t == 0) {
    uint32_t a = 0;
    for (int j = 0; j < 128; ++j) { uint32_t c = baseCnt[j]; baseCnt[j] = a; a += c; }
    baseCnt[128] = a;
  }
  __syncthreads();
  uint32_t base = baseCnt[t];
  for (int b = 0; b < 32; ++b) {
    if ((cw >> b) & 1u) {
      uint32_t rank = base + (uint32_t)__popc(cw & ((1u << b) - 1u));
      if (rank < MAXDET) {
        int i = t * 32 + b;
        out[rank * 4 + 0] = sBox[i * 4 + 0];
        out[rank * 4 + 1] = sBox[i * 4 + 1];
        out[rank * 4 + 2] = sBox[i * 4 + 2];
        out[rank * 4 + 3] = sBox[i * 4 + 3];
        out[4 * MAXDET + rank] = sScore[i];
        ((int*)out)[5 * MAXDET + rank] = sCls[i];
      }
    }
  }
  if (t == 0) {
    uint32_t tot = baseCnt[128];
    ((int*)out)[6 * MAXDET] = (int)(tot < (uint32_t)MAXDET ? tot : (uint32_t)MAXDET);
  }
}

// ---------------- host launcher ----------------
extern "C" void kernel_launch(void* const* d_in, const int* in_sizes, int n_in,
                              void* d_out, int out_size, void* d_ws, size_t ws_size,
                              hipStream_t stream) {
  const float* reg = (const float*)d_in[1];
  const float* cls = (const float*)d_in[2];
  const float* anc = (const float*)d_in[3];
  const int N = in_sizes[1] / 4;
  const int C = in_sizes[2] / N;
  const int img = (int)lround(sqrt((double)(in_sizes[0] / 3)));

  // workspace carve-up (all chunks 16B-aligned)
  char* p = (char*)d_ws;
  float*    wBoxes = (float*)p;    p += (size_t)N * 4 * sizeof(float);
  float*    wCand  = (float*)p;    p += (size_t)N * sizeof(float);
  int*      wCls   = (int*)p;      p += (size_t)N * sizeof(int);
  uint32_t* hist   = (uint32_t*)p; p += (size_t)NBINS * sizeof(uint32_t);
  uint32_t* scal   = (uint32_t*)p; p += 16 * sizeof(uint32_t);      // [0]=bstar [1]=counter
  uint32_t* selKey = (uint32_t*)p; p += (size_t)CAP * sizeof(uint32_t);
  int*      selIdx = (int*)p;      p += (size_t)CAP * sizeof(int);
  float*    sScore = (float*)p;    p += (size_t)TOPK * sizeof(float);
  float*    sBox   = (float*)p;    p += (size_t)TOPK * 4 * sizeof(float);
  int*      sCls   = (int*)p;      p += (size_t)TOPK * sizeof(int);
  uint32_t* sup    = (uint32_t*)p; p += (size_t)TOPK * (TOPK / 32) * sizeof(uint32_t);

  zero_u32<<<(NBINS + 16 + 255) / 256, 256, 0, stream>>>(hist, NBINS + 16); // hist + scal contiguous
  zero_u32<<<(out_size + 255) / 256, 256, 0, stream>>>((uint32_t*)d_out, out_size);

  k1_score_decode<<<(N + 127) / 128, 128, 0, stream>>>(cls, reg, anc, wBoxes, wCand, wCls,
                                                       N, C, (float)img, (float)img);
  k2a_hist<<<(N + 255) / 256, 256, 0, stream>>>(wCand, hist, N);
  k2b_thresh<<<1, 256, 0, stream>>>(hist, scal);
  k2c_collect<<<(N + 255) / 256, 256, 0, stream>>>(wCand, scal, &scal[1], selKey, selIdx, N);
  k2d_init<<<TOPK / 256, 256, 0, stream>>>(sScore, sBox, sCls);
  k2d_rank<<<CAP / 256, 256, 0, stream>>>(selKey, selIdx, scal, wBoxes, wCls, sScore, sBox, sCls);
  k3a_supmat<<<TOPK / 16, 32, 0, stream>>>(sBox, sup);
  k3b_nms<<<1, 128, 0, stream>>>(sup, sScore, sBox, sCls, (float*)d_out);
}